// MambaStock_88682484727910
// MI455X (gfx1250) — compile-verified
//
#include <hip/hip_runtime.h>
#include <cmath>

// ---------------- problem constants ----------------
#define L_SEQ   4096
#define DIN     8
#define DMODEL  256
#define NLAYERS 2
#define ED      512
#define NST     16
#define KCONV   4
#define DTRANK  16
#define EPSF    1e-5f
#define NCHUNK  64
#define CLEN    (L_SEQ / NCHUNK)   // 64

typedef __attribute__((ext_vector_type(2))) float v2f;
typedef __attribute__((ext_vector_type(8))) float v8f;

// Async global->LDS path (CDNA5 GLOBAL_LOAD_ASYNC_TO_LDS_B32, ASYNCcnt),
// guarded so a toolchain without the builtin still compiles via plain loads.
#if defined(__has_builtin)
#  if __has_builtin(__builtin_amdgcn_global_load_async_to_lds_b32)
#    define HAVE_ASYNC_LDS 1
#  endif
#  if __has_builtin(__builtin_amdgcn_s_wait_asynccnt)
#    define HAVE_WAIT_ASYNC_BUILTIN 1
#  endif
#endif

typedef __attribute__((address_space(1))) int* gptr_i32;
typedef __attribute__((address_space(3))) int* lptr_i32;

__device__ __forceinline__ void wait_asynccnt0() {
#if defined(HAVE_WAIT_ASYNC_BUILTIN)
    __builtin_amdgcn_s_wait_asynccnt(0);
#else
    asm volatile("s_wait_asynccnt 0" ::: "memory");
#endif
}

__device__ __forceinline__ float silu_f(float v) {
    return v / (1.0f + __expf(-v));
}

// ---------------- fc_in: h = x @ W^T + b  (L x 8 -> L x 256) ----------------
__global__ __launch_bounds__(256)
void fc_in_kernel(const float* __restrict__ x, const float* __restrict__ w,
                  const float* __restrict__ b, float* __restrict__ h) {
    int idx = blockIdx.x * 256 + threadIdx.x;     // over L*DMODEL
    int l = idx >> 8, m = idx & 255;
    float s = b[m];
    #pragma unroll
    for (int i = 0; i < DIN; ++i) s += x[l * DIN + i] * w[m * DIN + i];
    h[idx] = s;
}

// ---------------- rmsnorm over DMODEL=256 ----------------
__global__ __launch_bounds__(256)
void rmsnorm_kernel(const float* __restrict__ h, const float* __restrict__ w,
                    float* __restrict__ xn) {
    __shared__ float red[256];
    int l = blockIdx.x, t = threadIdx.x;
    float v = h[l * DMODEL + t];
    red[t] = v * v;
    __syncthreads();
    for (int s = 128; s > 0; s >>= 1) {
        if (t < s) red[t] += red[t + s];
        __syncthreads();
    }
    float scale = rsqrtf(red[0] * (1.0f / DMODEL) + EPSF);
    xn[l * DMODEL + t] = v * scale * w[t];
}

// ---------------- fp32 WMMA GEMM: C[M,N] = A[M,K] * Bw[N,K]^T ----------------
// One wave per 32x16 output tile: two 16x16 sub-tiles share each B fragment
// (halves B traffic, 2 wmma per loaded B). V_WMMA_F32_16X16X4_F32 over K.
// epilogue: 0 = none, 1 = softplus (after bias). bias/residual optional.
__global__ __launch_bounds__(32)
void gemm_wmma_f32(const float* __restrict__ A, int lda,
                   const float* __restrict__ Bw, int ldb,
                   float* __restrict__ Cout, int ldc, int Kdim,
                   const float* __restrict__ bias,
                   const float* __restrict__ residual, int ldr,
                   int epilogue) {
    const int m0 = blockIdx.x * 32;
    const int n0 = blockIdx.y * 16;
    const int lane = threadIdx.x;          // 0..31 (wave32)
    const int half = lane >> 4;            // 0 | 1
    const int mr   = lane & 15;
    const int koff = half * 2;

    const float* __restrict__ Arow0 = A  + (long)(m0 + mr) * lda;
    const float* __restrict__ Arow1 = A  + (long)(m0 + 16 + mr) * lda;
    const float* __restrict__ Brow  = Bw + (long)(n0 + mr) * ldb;

    v8f acc0 = {0.f, 0.f, 0.f, 0.f, 0.f, 0.f, 0.f, 0.f};
    v8f acc1 = {0.f, 0.f, 0.f, 0.f, 0.f, 0.f, 0.f, 0.f};
    for (int k = 0; k < Kdim; k += 4) {
        v2f b, a0, a1;
        b.x  = Brow[k + koff];
        b.y  = Brow[k + koff + 1];
        a0.x = Arow0[k + koff];
        a0.y = Arow0[k + koff + 1];
        a1.x = Arow1[k + koff];
        a1.y = Arow1[k + koff + 1];
        acc0 = __builtin_amdgcn_wmma_f32_16x16x4_f32(
            false, a0, false, b, (short)0, acc0, false, false);
        acc1 = __builtin_amdgcn_wmma_f32_16x16x4_f32(
            false, a1, false, b, (short)0, acc1, false, false);
    }

    const int col = n0 + mr;
    #pragma unroll
    for (int s = 0; s < 2; ++s) {
        v8f acc = s ? acc1 : acc0;
        int mbase = m0 + s * 16;
        #pragma unroll
        for (int i = 0; i < 8; ++i) {
            int row = mbase + i + half * 8;
            float v = acc[i];
            if (bias) v += bias[col];
            if (epilogue == 1) v = (v > 20.0f) ? v : log1pf(__expf(v));  // softplus
            if (residual) v += residual[(long)row * ldr + col];
            Cout[(long)row * ldc + col] = v;
        }
    }
}

// ---------------- causal depthwise conv (K=4) + silu ----------------
// xi = xz[:, :ED]; xc[l,e] = silu(sum_j w[e,j]*xi[l-3+j, e] + b[e])
__global__ __launch_bounds__(256)
void conv_silu_kernel(const float* __restrict__ xz, const float* __restrict__ cw,
                      const float* __restrict__ cb, float* __restrict__ xc) {
    int idx = blockIdx.x * 256 + threadIdx.x;   // over L*ED
    int l = idx >> 9, e = idx & (ED - 1);
    const float* w = cw + e * KCONV;
    float s = cb[e];
    #pragma unroll
    for (int j = 0; j < KCONV; ++j) {
        int t = l - (KCONV - 1) + j;
        if (t >= 0) s += w[j] * xz[t * (2 * ED) + e];
    }
    xc[idx] = silu_f(s);
}

// Preload this chunk's B/C slab (CLEN rows x 32 floats) from dbc into LDS.
// Uses async global->LDS when available; one barrier total for the chunk.
__device__ __forceinline__ void stage_bc_slab(const float* __restrict__ dbc,
                                              int t0, int e, float* sBC) {
#if defined(HAVE_ASYNC_LDS)
    #pragma unroll
    for (int i = 0; i < (CLEN * 32) / ED; ++i) {        // 4 elements/thread
        int o = e + i * ED;                             // 0..2047
        int t = o >> 5, q = o & 31;
        float* gp = (float*)(dbc + (long)(t0 + t) * 48 + DTRANK + q);  // drop const
        __builtin_amdgcn_global_load_async_to_lds_b32(
            (gptr_i32)gp, (lptr_i32)(sBC + o), 0, 0);
    }
    wait_asynccnt0();
#else
    #pragma unroll
    for (int i = 0; i < (CLEN * 32) / ED; ++i) {
        int o = e + i * ED;
        int t = o >> 5, q = o & 31;
        sBC[o] = dbc[(long)(t0 + t) * 48 + DTRANK + q];
    }
#endif
    __syncthreads();
}

// ---------------- scan pass 1: per-chunk transition (prod dA, local scan) ----------------
__global__ __launch_bounds__(512)
void scan_chunk_summary(const float* __restrict__ delta, const float* __restrict__ xc,
                        const float* __restrict__ dbc, const float* __restrict__ a_log,
                        float* __restrict__ Aprod, float* __restrict__ Bfin) {
    __shared__ float sBC[CLEN * 32];   // [t][0:16]=B, [t][16:32]=C (C unused here)
    int c = blockIdx.x;
    int e = threadIdx.x;
    int t0 = c * CLEN;
    stage_bc_slab(dbc, t0, e, sBC);

    float a[NST], hA[NST], hB[NST];
    #pragma unroll
    for (int n = 0; n < NST; ++n) {
        a[n]  = -__expf(a_log[e * NST + n]);
        hA[n] = 1.0f;
        hB[n] = 0.0f;
    }
    for (int tt = 0; tt < CLEN; ++tt) {
        int t = t0 + tt;
        float dl = delta[t * ED + e];
        float xv = xc[t * ED + e];
        float dx = dl * xv;
        const float* sB = sBC + tt * 32;
        #pragma unroll
        for (int n = 0; n < NST; ++n) {
            float da = __expf(dl * a[n]);
            hA[n] *= da;
            hB[n] = da * hB[n] + dx * sB[n];
        }
    }
    #pragma unroll
    for (int n = 0; n < NST; ++n) {
        Aprod[(c * ED + e) * NST + n] = hA[n];
        Bfin [(c * ED + e) * NST + n] = hB[n];
    }
}

// ---------------- scan pass 2: sequential combine of chunk summaries ----------------
__global__ __launch_bounds__(256)
void scan_combine(const float* __restrict__ Aprod, const float* __restrict__ Bfin,
                  float* __restrict__ Hinit) {
    int idx = blockIdx.x * 256 + threadIdx.x;   // over ED*NST = 8192
    float h = 0.0f;
    for (int c = 0; c < NCHUNK; ++c) {
        Hinit[c * (ED * NST) + idx] = h;
        h = Aprod[c * (ED * NST) + idx] * h + Bfin[c * (ED * NST) + idx];
    }
}

// ---------------- scan pass 3: replay chunk with correct h_init, emit y ----------------
__global__ __launch_bounds__(512)
void scan_emit(const float* __restrict__ delta, const float* __restrict__ xc,
               const float* __restrict__ dbc, const float* __restrict__ a_log,
               const float* __restrict__ Hinit, const float* __restrict__ Dw,
               const float* __restrict__ xz, float* __restrict__ y) {
    __shared__ float sBC[CLEN * 32];   // [t][0:16]=B, [t][16:32]=C
    int c = blockIdx.x;
    int e = threadIdx.x;
    int t0 = c * CLEN;
    stage_bc_slab(dbc, t0, e, sBC);

    float a[NST], h[NST];
    #pragma unroll
    for (int n = 0; n < NST; ++n) {
        a[n] = -__expf(a_log[e * NST + n]);
        h[n] = Hinit[c * (ED * NST) + e * NST + n];
    }
    float dcoef = Dw[e];
    for (int tt = 0; tt < CLEN; ++tt) {
        int t = t0 + tt;
        float dl = delta[t * ED + e];
        float xv = xc[t * ED + e];
        float dx = dl * xv;
        const float* sB = sBC + tt * 32;
        const float* sC = sB + NST;
        float yv = 0.0f;
        #pragma unroll
        for (int n = 0; n < NST; ++n) {
            float da = __expf(dl * a[n]);
            h[n] = da * h[n] + dx * sB[n];
            yv += h[n] * sC[n];
        }
        float zv = xz[t * (2 * ED) + ED + e];
        y[t * ED + e] = (yv + dcoef * xv) * silu_f(zv);
    }
}

// ---------------- fc_out + tanh: out[l] = tanh(h[l,:] . w + b) ----------------
__global__ __launch_bounds__(256)
void fc_out_kernel(const float* __restrict__ h, const float* __restrict__ w,
                   const float* __restrict__ b, float* __restrict__ out) {
    __shared__ float red[256];
    int l = blockIdx.x, t = threadIdx.x;
    red[t] = h[l * DMODEL + t] * w[t];
    __syncthreads();
    for (int s = 128; s > 0; s >>= 1) {
        if (t < s) red[t] += red[t + s];
        __syncthreads();
    }
    if (t == 0) out[l] = tanhf(red[0] + b[0]);
}

// ---------------- host orchestration ----------------
extern "C" void kernel_launch(void* const* d_in, const int* in_sizes, int n_in,
                              void* d_out, int out_size, void* d_ws, size_t ws_size,
                              hipStream_t stream) {
    const float* x        = (const float*)d_in[0];
    const float* fc_in_w  = (const float*)d_in[1];
    const float* fc_in_b  = (const float*)d_in[2];
    const float* fc_out_w = (const float*)d_in[3];
    const float* fc_out_b = (const float*)d_in[4];
    const float* norm_w   = (const float*)d_in[5];
    const float* in_w     = (const float*)d_in[6];
    const float* conv_w   = (const float*)d_in[7];
    const float* conv_b   = (const float*)d_in[8];
    const float* xp_w     = (const float*)d_in[9];
    const float* dt_w     = (const float*)d_in[10];
    const float* dt_b     = (const float*)d_in[11];
    const float* a_log    = (const float*)d_in[12];
    const float* Dw       = (const float*)d_in[13];
    const float* out_w    = (const float*)d_in[14];

    float* ws   = (float*)d_ws;
    float* h    = ws; ws += L_SEQ * DMODEL;        // residual stream
    float* xn   = ws; ws += L_SEQ * DMODEL;
    float* xz   = ws; ws += L_SEQ * 2 * ED;
    float* xc   = ws; ws += L_SEQ * ED;
    float* dbc  = ws; ws += L_SEQ * 48;
    float* delt = ws; ws += L_SEQ * ED;
    float* y    = ws; ws += L_SEQ * ED;
    float* Ap   = ws; ws += NCHUNK * ED * NST;
    float* Bf   = ws; ws += NCHUNK * ED * NST;
    float* Hi   = ws; ws += NCHUNK * ED * NST;

    fc_in_kernel<<<L_SEQ, 256, 0, stream>>>(x, fc_in_w, fc_in_b, h);

    for (int lyr = 0; lyr < NLAYERS; ++lyr) {
        rmsnorm_kernel<<<L_SEQ, 256, 0, stream>>>(h, norm_w + lyr * DMODEL, xn);

        // in_proj: xz = xn @ in_w^T   (4096 x 1024, K=256)
        dim3 g1(L_SEQ / 32, (2 * ED) / 16);
        gemm_wmma_f32<<<g1, 32, 0, stream>>>(xn, DMODEL,
                                             in_w + (long)lyr * 2 * ED * DMODEL, DMODEL,
                                             xz, 2 * ED, DMODEL,
                                             nullptr, nullptr, 0, 0);

        conv_silu_kernel<<<(L_SEQ * ED) / 256, 256, 0, stream>>>(
            xz, conv_w + lyr * ED * KCONV, conv_b + lyr * ED, xc);

        // x_proj: dbc = xc @ xp_w^T   (4096 x 48, K=512)
        dim3 g2(L_SEQ / 32, 48 / 16);
        gemm_wmma_f32<<<g2, 32, 0, stream>>>(xc, ED,
                                             xp_w + (long)lyr * 48 * ED, ED,
                                             dbc, 48, ED,
                                             nullptr, nullptr, 0, 0);

        // dt_proj + bias + softplus: delta = softplus(dbc[:, :16] @ dt_w^T + dt_b)
        dim3 g3(L_SEQ / 32, ED / 16);
        gemm_wmma_f32<<<g3, 32, 0, stream>>>(dbc, 48,
                                             dt_w + (long)lyr * ED * DTRANK, DTRANK,
                                             delt, ED, DTRANK,
                                             dt_b + lyr * ED, nullptr, 0, /*softplus*/1);

        // chunked parallel scan (3 passes)
        scan_chunk_summary<<<NCHUNK, ED, 0, stream>>>(delt, xc, dbc,
                                                      a_log + lyr * ED * NST, Ap, Bf);
        scan_combine<<<(ED * NST) / 256, 256, 0, stream>>>(Ap, Bf, Hi);
        scan_emit<<<NCHUNK, ED, 0, stream>>>(delt, xc, dbc,
                                             a_log + lyr * ED * NST, Hi,
                                             Dw + lyr * ED, xz, y);

        // out_proj + residual: h = h + y @ out_w^T  (4096 x 256, K=512)
        dim3 g4(L_SEQ / 32, DMODEL / 16);
        gemm_wmma_f32<<<g4, 32, 0, stream>>>(y, ED,
                                             out_w + (long)lyr * DMODEL * ED, ED,
                                             h, DMODEL, ED,
                                             nullptr, h, DMODEL, 0);
    }

    fc_out_kernel<<<L_SEQ, 256, 0, stream>>>(h, fc_out_w, fc_out_b, (float*)d_out);
}